// Electron_Density_Descriptor_89979564851607
// MI455X (gfx1250) — compile-verified
//
#include <hip/hip_runtime.h>

// Problem constants from the reference
#define NATOMS 1024
#define TOPK   16
#define UCENT  10
#define SGRID  100
#define CCH    8
#define L2DIM  16
#define OUTCH  8

typedef __attribute__((ext_vector_type(2))) float v2f;
typedef __attribute__((ext_vector_type(8))) float v8f;

__global__ __launch_bounds__(256)
void edd_kernel(const float* __restrict__ rji,    // (N,K,3)
                const float* __restrict__ sigma,  // (N,K,U,C)
                const float* __restrict__ co,     // (N,K,U,C)
                const float* __restrict__ mask,   // (N,K)
                const float* __restrict__ gama,   // (U)
                const float* __restrict__ grid,   // (S,3)
                const float* __restrict__ Ylm,    // (L2,S)
                const float* __restrict__ w,      // (LMAX+1,OUTC,C)
                const float* __restrict__ bias,   // (OUTC)
                float* __restrict__ out)          // (N,L2,OUTC)
{
    __shared__ float s_gx[SGRID], s_gy[SGRID], s_gz[SGRID];
    __shared__ float s_rji[TOPK][3];
    __shared__ float s_mask[TOPK];
    __shared__ float s_gama[UCENT];
    // atom_sph, channel-major and padded to 16 rows so the WMMA B-operand
    // can be fed with unconditional, full-EXEC ds_load_b64 (rows 8..15 = 0).
    __shared__ float s_sphT[16][SGRID];
    // padded to 16 columns so the D-matrix store needs no lane predication
    __shared__ float s_proj[L2DIM][16];

    const int n   = blockIdx.x;
    const int tid = threadIdx.x;

    // ---- stage per-atom + constant data into LDS; zero the B padding rows ----
    if (tid < SGRID) {
        s_gx[tid] = grid[tid * 3 + 0];
        s_gy[tid] = grid[tid * 3 + 1];
        s_gz[tid] = grid[tid * 3 + 2];
    }
    if (tid < TOPK) {
        const int b = (n * TOPK + tid) * 3;
        s_rji[tid][0] = rji[b + 0];
        s_rji[tid][1] = rji[b + 1];
        s_rji[tid][2] = rji[b + 2];
        s_mask[tid]   = mask[n * TOPK + tid];
    }
    if (tid < UCENT) s_gama[tid] = gama[tid];
    {   // zero rows 8..15 of s_sphT (8*100 = 800 floats)
        float* pad = &s_sphT[CCH][0];
        pad[tid]       = 0.f;
        pad[tid + 256] = 0.f;
        pad[tid + 512] = 0.f;
        if (tid < 800 - 768) pad[tid + 768] = 0.f;
    }
    __syncthreads();

    // ---- Phase 1: density on sphere grid (exp-bound) ----
    // thread -> (c, s0); each thread owns s0, s0+32, s0+64, (s0+96)
    const int c  = tid & (CCH - 1);
    const int s0 = tid >> 3;                  // 0..31
    const bool has3 = (s0 + 96) < SGRID;      // s0 < 4

    const float gx0 = s_gx[s0],      gy0 = s_gy[s0],      gz0 = s_gz[s0];
    const float gx1 = s_gx[s0 + 32], gy1 = s_gy[s0 + 32], gz1 = s_gz[s0 + 32];
    const float gx2 = s_gx[s0 + 64], gy2 = s_gy[s0 + 64], gz2 = s_gz[s0 + 64];
    const float gx3 = has3 ? s_gx[s0 + 96] : 0.f;
    const float gy3 = has3 ? s_gy[s0 + 96] : 0.f;
    const float gz3 = has3 ? s_gz[s0 + 96] : 0.f;

    float acc0 = 0.f, acc1 = 0.f, acc2 = 0.f, acc3 = 0.f;

    const float* sig_n = sigma + (size_t)n * TOPK * UCENT * CCH;
    const float* co_n  = co    + (size_t)n * TOPK * UCENT * CCH;

    for (int k = 0; k < TOPK; ++k) {
        const float rx = s_rji[k][0], ry = s_rji[k][1], rz = s_rji[k][2];
        const float mk = s_mask[k];
        #pragma unroll
        for (int u = 0; u < UCENT; ++u) {
            const float ga = s_gama[u];
            const float cx = ga * rx, cy = ga * ry, cz = ga * rz;
            const int idx = (k * UCENT + u) * CCH + c;
            const float sg = fabsf(sig_n[idx]);
            const float cm = co_n[idx] * mk;

            float dx = gx0 - cx, dy = gy0 - cy, dz = gz0 - cz;
            acc0 += cm * __expf(-(dx*dx + dy*dy + dz*dz) * sg);
            dx = gx1 - cx; dy = gy1 - cy; dz = gz1 - cz;
            acc1 += cm * __expf(-(dx*dx + dy*dy + dz*dz) * sg);
            dx = gx2 - cx; dy = gy2 - cy; dz = gz2 - cz;
            acc2 += cm * __expf(-(dx*dx + dy*dy + dz*dz) * sg);
            if (has3) {
                dx = gx3 - cx; dy = gy3 - cy; dz = gz3 - cz;
                acc3 += cm * __expf(-(dx*dx + dy*dy + dz*dz) * sg);
            }
        }
    }
    s_sphT[c][s0]      = acc0;
    s_sphT[c][s0 + 32] = acc1;
    s_sphT[c][s0 + 64] = acc2;
    if (has3) s_sphT[c][s0 + 96] = acc3;
    __syncthreads();

    // ---- Phase 2: proj = (1/S) * Ylm(16x100) x atom_sph(100x8), via WMMA ----
    if (tid < 32) {
#if defined(__HIP_DEVICE_COMPILE__) && defined(__gfx1250__)
        const int lane = tid;
        const int half = lane >> 4;   // lanes 16..31 handle K+2, K+3
        const int mr   = lane & 15;   // A row (M) / B column (N)
        v8f accw = {};
        #pragma unroll
        for (int kk = 0; kk < SGRID; kk += 4) {
            const int kb = kk + 2 * half;     // even -> 8B-aligned pairs
            // A (Ylm): 16x4 f32 tile; VGPR0={K,K+2}, VGPR1={K+1,K+3}, M = lane%16
            v2f a = *(const v2f*)&Ylm[mr * SGRID + kb];
            // B (atom_sph^T): padded rows make this unconditional ds_load_b64
            v2f b = *(const v2f*)&s_sphT[mr][kb];
            accw = __builtin_amdgcn_wmma_f32_16x16x4_f32(
                false, a, false, b, (short)0, accw, false, false);
        }
        const float invS = 1.0f / (float)SGRID;
        #pragma unroll
        for (int j = 0; j < 8; ++j) {
            // C/D layout: VGPR j, lanes 0-15 -> M=j; lanes 16-31 -> M=j+8
            const int m = j + 8 * half;
            s_proj[m][mr] = accw[j] * invS;   // cols >= CCH are dead padding
        }
#else
        // host-pass / non-gfx1250 fallback (never executed on MI455X)
        for (int l = tid; l < L2DIM; l += 32) {
            for (int cc = 0; cc < CCH; ++cc) {
                float sacc = 0.f;
                for (int ss = 0; ss < SGRID; ++ss)
                    sacc += Ylm[l * SGRID + ss] * s_sphT[cc][ss];
                s_proj[l][cc] = sacc / (float)SGRID;
            }
        }
#endif
    }
    __syncthreads();

    // ---- Phase 3: SO3 linear (l-dependent weights -> VALU dots) ----
    if (tid < L2DIM * OUTCH) {
        const int l = tid >> 3;       // OUTCH == 8
        const int o = tid & 7;
        const int lq = (l == 0) ? 0 : (l < 4) ? 1 : (l < 9) ? 2 : 3;
        const float* wl = w + (lq * OUTCH + o) * CCH;
        float sum = 0.f;
        #pragma unroll
        for (int cc = 0; cc < CCH; ++cc) sum += s_proj[l][cc] * wl[cc];
        if (l == 0) sum += bias[o];
        out[((size_t)n * L2DIM + l) * OUTCH + o] = sum;
    }
}

extern "C" void kernel_launch(void* const* d_in, const int* in_sizes, int n_in,
                              void* d_out, int out_size, void* d_ws, size_t ws_size,
                              hipStream_t stream) {
    // setup_inputs order:
    // 0: atom_positions (unused: cancels algebraically)
    // 1: rji  2: sigma  3: co  4: neighbor_mask  5: gama
    // 6: sphere_grid  7: Y_lm  8: so3_weight  9: so3_bias
    const float* rji   = (const float*)d_in[1];
    const float* sigma = (const float*)d_in[2];
    const float* co    = (const float*)d_in[3];
    const float* nmask = (const float*)d_in[4];
    const float* gama  = (const float*)d_in[5];
    const float* grid  = (const float*)d_in[6];
    const float* Ylm   = (const float*)d_in[7];
    const float* w     = (const float*)d_in[8];
    const float* bias  = (const float*)d_in[9];
    float* out = (float*)d_out;

    edd_kernel<<<NATOMS, 256, 0, stream>>>(rji, sigma, co, nmask, gama, grid,
                                           Ylm, w, bias, out);
}